// SSIMLoss_3607772528993
// MI455X (gfx1250) — compile-verified
//
#include <hip/hip_runtime.h>

// SSIM loss on MI455X (gfx1250, wave32).
// Stage 1: one wave per 16x16 tile. Vertical 11-tap box sums via running sums
// (lane-per-column), horizontal 11-tap box sums via V_WMMA_F32_16X16X4_F32
// against a constant 26x16 band matrix of ones. Fused SSIM + reduction.
// Stage 2: deterministic single-block reduction of per-block partials.

typedef __attribute__((ext_vector_type(2))) float v2f;
typedef __attribute__((ext_vector_type(8))) float v8f;

#define HW      512
#define TILE    16
#define HALO    26      // TILE + 2*5
#define STRIDE  28      // LDS row stride (floats); 28 mod 64 banks -> conflict free
#define WAVES   4       // waves (=tiles) per block
// per-wave LDS floats: xt(26*28=728) + yt(728) + vs(5*16*28=2240) = 3696
#define WAVE_LDS 3696
#define NTILES  (64 * 32 * 32)          // 65536
#define NBLOCKS (NTILES / WAVES)        // 16384

__global__ __launch_bounds__(128)
void ssim_stage1(const float* __restrict__ x, const float* __restrict__ y,
                 float* __restrict__ partials)
{
    __shared__ float smem[WAVES * WAVE_LDS];
    __shared__ float wsum[WAVES];

    const int lane = threadIdx.x & 31;
    const int wave = threadIdx.x >> 5;
    float* base = smem + wave * WAVE_LDS;
    float* xt = base;                 // 26 x 26 halo of x (stride 28)
    float* yt = base + 728;           // 26 x 26 halo of y
    float* vs = base + 1456;          // 5 quantities x 16 x 26 vertical sums

    const int tile = blockIdx.x * WAVES + wave;   // exact: 16384*4 == 65536
    const int img  = tile >> 10;                  // tile / 1024
    const int tin  = tile & 1023;
    const int tr   = tin >> 5;
    const int tc   = tin & 31;
    const int r0   = tr * TILE - 5;
    const int c0   = tc * TILE - 5;
    const float* xi = x + (size_t)img * HW * HW;
    const float* yi = y + (size_t)img * HW * HW;

    // ---- Phase 1: global -> LDS halo load, zero padded (matches conv zero-pad)
    for (int i = lane; i < HALO * HALO; i += 32) {
        int r = i / HALO;
        int c = i - r * HALO;
        int gr = r0 + r, gc = c0 + c;
        bool ok = (gr >= 0) & (gr < HW) & (gc >= 0) & (gc < HW);
        int gi = gr * HW + gc;
        float xv = ok ? xi[gi] : 0.0f;
        float yv = ok ? yi[gi] : 0.0f;
        xt[r * STRIDE + c] = xv;
        yt[r * STRIDE + c] = yv;
    }
    __syncthreads();

    // ---- Phase 2: vertical 11-tap running sums; lane = halo column (26 of 32 lanes)
    if (lane < HALO) {
        const int c = lane;
        float sx = 0.f, sy = 0.f, sxx = 0.f, syy = 0.f, sxy = 0.f;
        #pragma unroll
        for (int r = 0; r < 11; ++r) {
            float xv = xt[r * STRIDE + c], yv = yt[r * STRIDE + c];
            sx += xv; sy += yv; sxx += xv * xv; syy += yv * yv; sxy += xv * yv;
        }
        vs[0 * 448 + c] = sx;
        vs[1 * 448 + c] = sy;
        vs[2 * 448 + c] = sxx;
        vs[3 * 448 + c] = syy;
        vs[4 * 448 + c] = sxy;
        for (int m = 1; m < 16; ++m) {
            float xn = xt[(m + 10) * STRIDE + c], yn = yt[(m + 10) * STRIDE + c];
            float xo = xt[(m - 1) * STRIDE + c],  yo = yt[(m - 1) * STRIDE + c];
            sx  += xn - xo;
            sy  += yn - yo;
            sxx += xn * xn - xo * xo;
            syy += yn * yn - yo * yo;
            sxy += xn * yn - xo * yo;
            vs[0 * 448 + m * STRIDE + c] = sx;
            vs[1 * 448 + m * STRIDE + c] = sy;
            vs[2 * 448 + m * STRIDE + c] = sxx;
            vs[3 * 448 + m * STRIDE + c] = syy;
            vs[4 * 448 + m * STRIDE + c] = sxy;
        }
    }
    __syncthreads();

    // ---- Phase 3: horizontal 11-tap box sums as WMMA against constant band.
    // out(16x16) = VS(16x26) * B(26x16), B[k][n] = (n <= k <= n+10) ? 1 : 0.
    // A layout (f32 16x4): lanes 0-15 -> M=0..15, VGPR0=K0, VGPR1=K1;
    //                      lanes 16-31 -> same M, VGPR0=K2, VGPR1=K3.
    v8f acc[5];
    #pragma unroll
    for (int q = 0; q < 5; ++q)
        #pragma unroll
        for (int e = 0; e < 8; ++e) acc[q][e] = 0.0f;

    const int mn   = lane & 15;                 // M for A, N for B
    const int ksel = (lane < 16) ? 0 : 2;
    #pragma unroll
    for (int q = 0; q < 5; ++q) {
        const float* vq = vs + q * 448;
        #pragma unroll
        for (int kk = 0; kk < 7; ++kk) {        // K padded 26 -> 28
            int k0 = kk * 4 + ksel;
            int k1 = k0 + 1;
            int k0c = (k0 < HALO - 1) ? k0 : (HALO - 1);
            int k1c = (k1 < HALO - 1) ? k1 : (HALO - 1);
            float a0 = vq[mn * STRIDE + k0c];
            float a1 = vq[mn * STRIDE + k1c];
            v2f a, b;
            a.x = (k0 < HALO) ? a0 : 0.0f;
            a.y = (k1 < HALO) ? a1 : 0.0f;
            b.x = (k0 >= mn && k0 <= mn + 10) ? 1.0f : 0.0f;
            b.y = (k1 >= mn && k1 <= mn + 10) ? 1.0f : 0.0f;
            acc[q] = __builtin_amdgcn_wmma_f32_16x16x4_f32(
                false, a, false, b, (short)0, acc[q], false, false);
        }
    }

    // ---- Phase 4: SSIM map + reduction. acc element e -> (M = e + (lane<16?0:8), N = lane&15)
    const float inv121 = 1.0f / 121.0f;
    const float C1 = 1.0e-4f;   // 0.01^2
    const float C2 = 9.0e-4f;   // 0.03^2
    float lsum = 0.0f;
    #pragma unroll
    for (int e = 0; e < 8; ++e) {
        float mx  = acc[0][e] * inv121;
        float my  = acc[1][e] * inv121;
        float exx = acc[2][e] * inv121;
        float eyy = acc[3][e] * inv121;
        float exy = acc[4][e] * inv121;
        float mx2 = mx * mx, my2 = my * my, mxy = mx * my;
        float sx2 = exx - mx2, sy2 = eyy - my2, sxy = exy - mxy;
        float num = (2.0f * mxy + C1) * (2.0f * sxy + C2);
        float den = (mx2 + my2 + C1) * (sx2 + sy2 + C2);
        lsum += num / den;
    }
    // wave32 reduction
    #pragma unroll
    for (int off = 16; off > 0; off >>= 1)
        lsum += __shfl_down(lsum, off, 32);
    if (lane == 0) wsum[wave] = lsum;
    __syncthreads();
    if (threadIdx.x == 0) {
        float t = 0.0f;
        #pragma unroll
        for (int w = 0; w < WAVES; ++w) t += wsum[w];
        partials[blockIdx.x] = t;
    }
}

__global__ __launch_bounds__(256)
void ssim_stage2(const float* __restrict__ partials, int n, float* __restrict__ out)
{
    __shared__ double sm[256];
    double s = 0.0;
    for (int i = threadIdx.x; i < n; i += 256) s += (double)partials[i];
    sm[threadIdx.x] = s;
    __syncthreads();
    for (int off = 128; off > 0; off >>= 1) {
        if ((int)threadIdx.x < off) sm[threadIdx.x] += sm[threadIdx.x + off];
        __syncthreads();
    }
    if (threadIdx.x == 0) {
        double mean = sm[0] / (double)(64.0 * 512.0 * 512.0);
        out[0] = (float)(1.0 - mean);
    }
}

extern "C" void kernel_launch(void* const* d_in, const int* in_sizes, int n_in,
                              void* d_out, int out_size, void* d_ws, size_t ws_size,
                              hipStream_t stream)
{
    const float* x = (const float*)d_in[0];
    const float* y = (const float*)d_in[1];
    // d_in[2] is the box kernel (ones / 121); its value is baked in (1/121 scaling).
    float* partials = (float*)d_ws;                  // NBLOCKS floats = 64 KB
    ssim_stage1<<<NBLOCKS, 128, 0, stream>>>(x, y, partials);
    ssim_stage2<<<1, 256, 0, stream>>>(partials, NBLOCKS, (float*)d_out);
}